// SeesawLossWithLogits_73289321939313
// MI455X (gfx1250) — compile-verified
//
#include <hip/hip_runtime.h>
#include <math.h>

// Seesaw loss, reduced form: only sigma[b, t_b] contributes, so the [B,C]x[C,C]
// GEMM collapses to one dot product per batch row. Bandwidth-bound:
//   logits 105 MB @ 23.3 TB/s  ->  ~4.5 us floor (s fits in 192 MB L2).
// Per-WG: 16 batch rows -> 16x16xC f32 WMMA tile (V_WMMA_F32_16X16X4_F32),
// diagonal kept. K split across 8 waves, fully unrolled scalar control flow.

#define BATCH   16384
#define NCLS    1604
#define TILE_B  16
#define CHUNK   256          // K staging chunk; NCLS = 6*256 + 68
#define NFULL   6
#define KTAIL   68           // tail chunk, 17 k-steps of 4
#define LSTRIDE (CHUNK + 4)  // pad: lane m -> bank 4m, conflict-free b64 frag loads
#define EPSF    1e-6f

typedef __attribute__((ext_vector_type(2))) float v2f;
typedef __attribute__((ext_vector_type(8))) float v8f;

__global__ void seesaw_zero_out(float* out) {
    if (threadIdx.x == 0) *out = 0.0f;
}

__device__ __forceinline__ void stage_chunk(
    const float* __restrict__ logits, const float* __restrict__ s,
    float (*sE)[LSTRIDE], float (*sS)[LSTRIDE], const int* tgt,
    const float* rowmax, int b0, int kb, int sw, int lane, int kc4)
{
    #pragma unroll
    for (int rr = 0; rr < 2; ++rr) {
        const int r = sw + rr * 8;
        const float4* lrow = (const float4*)(logits + (size_t)(b0 + r) * NCLS + kb);
        const float4* srow = (const float4*)(s + (size_t)tgt[r] * NCLS + kb);
        const float mx = rowmax[r];
        for (int c4 = lane; c4 < kc4; c4 += 32) {
            float4 lv = lrow[c4];
            float4 sv = srow[c4];
            float4 ev;
            ev.x = __expf(lv.x - mx);
            ev.y = __expf(lv.y - mx);
            ev.z = __expf(lv.z - mx);
            ev.w = __expf(lv.w - mx);
            *(float4*)&sE[r][c4 * 4] = ev;
            *(float4*)&sS[r][c4 * 4] = sv;
        }
    }
}

__global__ __launch_bounds__(256) void seesaw_loss_kernel(
    const float* __restrict__ logits,   // [BATCH, NCLS]
    const float* __restrict__ s,        // [NCLS, NCLS]
    const int*   __restrict__ targets,  // [BATCH]
    float* __restrict__ out)            // [1], pre-zeroed
{
    __shared__ float sE[TILE_B][LSTRIDE];   // exp(logits - rowmax), K-chunk
    __shared__ float sS[TILE_B][LSTRIDE];   // gathered s[t_n, :] rows, K-chunk
    __shared__ float red[8 * 256];          // 8 per-wave accumulator fragments
    __shared__ float Dtile[TILE_B * TILE_B];
    __shared__ float rowmax[TILE_B];
    __shared__ float selfnum[TILE_B];
    __shared__ float selfs[TILE_B];
    __shared__ int   tgt[TILE_B];

    const int tid  = threadIdx.x;
    const int lane = tid & 31;
    // Wave id as an SGPR: gives scalar (s_cbranch) control flow around WMMA
    // blocks, so EXEC stays all-1s at every WMMA as the ISA requires.
    const int sw   = __builtin_amdgcn_readfirstlane(tid >> 5);   // 0..7
    const int b0   = blockIdx.x * TILE_B;

    if (tid < TILE_B) tgt[tid] = targets[b0 + tid];
    __syncthreads();

    // ---- Pass 1: row maxes (wave w owns rows w and w+8), logits from HBM ----
    #pragma unroll
    for (int rr = 0; rr < 2; ++rr) {
        const int r = sw + rr * 8;
        const float4* row = (const float4*)(logits + (size_t)(b0 + r) * NCLS);
        float m = -3.0e38f;
        for (int c4 = lane; c4 < NCLS / 4; c4 += 32) {
            float4 v = row[c4];
            m = fmaxf(m, fmaxf(fmaxf(v.x, v.y), fmaxf(v.z, v.w)));
        }
        #pragma unroll
        for (int off = 16; off > 0; off >>= 1)
            m = fmaxf(m, __shfl_xor(m, off, 32));
        if (lane == 0) rowmax[r] = m;
    }
    __syncthreads();

    if (tid < TILE_B) {
        const int t = tgt[tid];
        selfnum[tid] = __expf(logits[(size_t)(b0 + tid) * NCLS + t] - rowmax[tid]);
        selfs[tid]   = s[(size_t)t * NCLS + t];
    }

    // ---- Pass 2: chunked exp + gather, WMMA accumulate (re-reads hit L2) ----
    v8f acc = {};   // persistent 16x16 f32 accumulator fragment per wave

    // Fragment addressing (32-bit WMMA striping):
    //   A: VGPR0 = K0 (lanes 0-15) | K2 (lanes 16-31), VGPR1 = K1 | K3.
    //   B mirrored. Lane's M (for A) == lane's N (for B) == lane & 15.
    const int mn = lane & 15;
    const int hk = (lane >> 4) << 1;         // +0 (lo half) or +2 (hi half)
    const int kbase = 4 * sw + hk;           // this wave's first k-step offset

    // 6 full chunks: 64 k-steps each, 8 per wave, fully unrolled.
    for (int cb = 0; cb < NFULL; ++cb) {
        const int kb = cb * CHUNK;
        __syncthreads();                      // prior chunk's WMMA reads done
        stage_chunk(logits, s, sE, sS, tgt, rowmax, b0, kb, sw, lane, CHUNK / 4);
        __syncthreads();
        #pragma unroll
        for (int u = 0; u < 8; ++u) {        // ks = sw + 8u -> k0 = 4sw+32u+hk
            const int k0 = kbase + 32 * u;
            v2f a = *(const v2f*)&sE[mn][k0];
            v2f b = *(const v2f*)&sS[mn][k0];
            acc = __builtin_amdgcn_wmma_f32_16x16x4_f32(
                false, a, false, b, (short)0, acc, false, false);
        }
    }

    // Tail chunk: 68 cols = 17 k-steps. Every wave takes ks = sw, sw+8;
    // wave 0 additionally takes ks = 16 (scalar branch, EXEC stays full).
    {
        const int kb = NFULL * CHUNK;
        __syncthreads();
        stage_chunk(logits, s, sE, sS, tgt, rowmax, b0, kb, sw, lane, KTAIL / 4);
        __syncthreads();
        #pragma unroll
        for (int u = 0; u < 2; ++u) {
            const int k0 = kbase + 32 * u;
            v2f a = *(const v2f*)&sE[mn][k0];
            v2f b = *(const v2f*)&sS[mn][k0];
            acc = __builtin_amdgcn_wmma_f32_16x16x4_f32(
                false, a, false, b, (short)0, acc, false, false);
        }
        if (sw == 0) {
            const int k0 = 64 + hk;          // ks = 16
            v2f a = *(const v2f*)&sE[mn][k0];
            v2f b = *(const v2f*)&sS[mn][k0];
            acc = __builtin_amdgcn_wmma_f32_16x16x4_f32(
                false, a, false, b, (short)0, acc, false, false);
        }
    }
    __syncthreads();

    // ---- Cross-wave reduction of the 8 accumulator fragments ----
    #pragma unroll
    for (int r = 0; r < 8; ++r) red[sw * 256 + r * 32 + lane] = acc[r];
    __syncthreads();
    {
        // wave w reduces fragment-VGPR slot r == w across all 8 waves
        float v = 0.0f;
        #pragma unroll
        for (int w2 = 0; w2 < 8; ++w2) v += red[w2 * 256 + sw * 32 + lane];
        const int m = sw + 8 * (lane >> 4);  // C/D layout: VGPR r, lane half
        const int n = lane & 15;
        Dtile[m * 16 + n] = v;
    }
    __syncthreads();

    // ---- Final per-row loss from the diagonal; block-partial into d_out ----
    if (tid == 0) {
        float sum = 0.0f;
        for (int i = 0; i < TILE_B; ++i) {
            const float num   = selfnum[i];
            const float denom = Dtile[i * 16 + i] - selfs[i] * num + num;
            const float sigma = num / (denom + EPSF);
            sum += -__logf(sigma + EPSF);
        }
        atomicAdd(out, sum * (1.0f / (float)BATCH));
    }
}

extern "C" void kernel_launch(void* const* d_in, const int* in_sizes, int n_in,
                              void* d_out, int out_size, void* d_ws, size_t ws_size,
                              hipStream_t stream) {
    (void)in_sizes; (void)n_in; (void)out_size; (void)d_ws; (void)ws_size;
    const float* logits  = (const float*)d_in[0];
    const float* s       = (const float*)d_in[1];
    const int*   targets = (const int*)d_in[2];
    float*       out     = (float*)d_out;

    seesaw_zero_out<<<1, 32, 0, stream>>>(out);
    seesaw_loss_kernel<<<BATCH / TILE_B, 256, 0, stream>>>(logits, s, targets, out);
}